// EarlyFusionModel_56023553409154
// MI455X (gfx1250) — compile-verified
//
#include <hip/hip_runtime.h>
#include <stdint.h>

// ---------------------------------------------------------------------------
// EarlyFusionModel 'match_dates' fusion for MI455X (gfx1250).
// Pure bandwidth problem (~390 MB traffic, 0 FLOPs -> ~17us at 23.3 TB/s).
// Sat-channel planes are contiguous 64KB copies -> use the CDNA5 async
// global<->LDS path (ASYNCcnt). Climate channels are scalar splats -> NT
// b128 stores.
// ---------------------------------------------------------------------------

#define PAD_VALUE (-1000.0f)

typedef float v4f __attribute__((ext_vector_type(4)));

constexpr int kB    = 8;
constexpr int kT    = 24;
constexpr int kC    = 10;     // sat channels
constexpr int kCc   = 11;     // climate channels
constexpr int kCout = kC + kCc;   // 21
constexpr int kH    = 128;
constexpr int kW    = 128;
constexpr int kHW   = kH * kW;          // 16384 floats
constexpr int kPlaneBytes = kHW * 4;    // 65536 bytes per (b,t,ch) plane
constexpr int kTclim = 365;

// ---------------------------------------------------------------------------
// Kernel 1: per-(b,t) date match + gather of the climate vector.
//   clim_vec[bt*kCc + cc] = has_match ? input_clim[b, idx, cc] : PAD_VALUE
// ---------------------------------------------------------------------------
__global__ void clim_match_kernel(const int* __restrict__ dates_sat,     // [B*T]
                                  const float* __restrict__ input_clim,  // [B*365*Cc]
                                  const int* __restrict__ dates_clim,    // [B*365]
                                  float* __restrict__ clim_vec)          // [B*T*Cc]
{
    int i = blockIdx.x * blockDim.x + threadIdx.x;   // flat (b*T + t)
    if (i >= kB * kT) return;
    int b = i / kT;
    int d = dates_sat[i];
    int idx = 0;
    bool found = false;
    const int* dc = dates_clim + b * kTclim;
    for (int j = 0; j < kTclim; ++j) {
        // first match (argmax-of-bool semantics)
        if (!found && dc[j] == d) { idx = j; found = true; }
    }
    const float* src = input_clim + ((size_t)b * kTclim + idx) * kCc;
    float* dst = clim_vec + (size_t)i * kCc;
    #pragma unroll
    for (int c = 0; c < kCc; ++c)
        dst[c] = found ? src[c] : PAD_VALUE;
}

// ---------------------------------------------------------------------------
// Kernel 2: copy the 1920 sat planes through LDS with CDNA5 async copies.
// One block per plane (256 threads = 8 waves). Each wave owns a 2KB LDS slot
// and streams 4 segments of 2KB: 4x async b128 loads (512B each across 32
// lanes), wait, 4x async b128 stores, wait. Loop bounds are wave-uniform so
// EXEC is all-ones for every async op.
// ---------------------------------------------------------------------------
__global__ __launch_bounds__(256)
void sat_copy_async_kernel(const float* __restrict__ sat,   // [B*T*C*HW]
                           float* __restrict__ out)         // [B*T*Cout*HW]
{
    __shared__ __align__(16) char smem[8 * 2048];

    const int p    = blockIdx.x;           // plane 0..B*T*C-1
    const int lane = threadIdx.x & 31;
    const int wave = threadIdx.x >> 5;     // 0..7

    const int bt = p / kC;
    const int ch = p - bt * kC;

    const char* src_plane = (const char*)sat + (size_t)p * kPlaneBytes;
    char*       dst_plane = (char*)out + ((size_t)bt * kCout + ch) * (size_t)kPlaneBytes;

    // Workgroup-relative LDS byte address: low 32 bits of the generic pointer
    // to a __shared__ object are exactly the LDS offset (LDS aperture mapping).
    unsigned lds_addr = (unsigned)(uintptr_t)(&smem[0]) + (unsigned)(wave * 2048 + lane * 16);

    // 32 segments of 2KB per 64KB plane; wave w handles w, w+8, w+16, w+24.
    for (int seg = wave; seg < 32; seg += 8) {
        const char* s = src_plane + seg * 2048 + lane * 16;
        char*       d = dst_plane + seg * 2048 + lane * 16;
        asm volatile(
            "global_load_async_to_lds_b128 %0, %1, off\n\t"
            "global_load_async_to_lds_b128 %0, %1, off offset:512\n\t"
            "global_load_async_to_lds_b128 %0, %1, off offset:1024\n\t"
            "global_load_async_to_lds_b128 %0, %1, off offset:1536\n\t"
            "s_wait_asynccnt 0x0\n\t"
            "global_store_async_from_lds_b128 %2, %0, off\n\t"
            "global_store_async_from_lds_b128 %2, %0, off offset:512\n\t"
            "global_store_async_from_lds_b128 %2, %0, off offset:1024\n\t"
            "global_store_async_from_lds_b128 %2, %0, off offset:1536\n\t"
            "s_wait_asynccnt 0x0"
            :
            : "v"(lds_addr), "v"(s), "v"(d)
            : "memory");
    }
}

// ---------------------------------------------------------------------------
// Kernel 3: broadcast each matched climate scalar over its 64KB output plane.
// grid = (HW/4/256, Cc, B*T); one v4f NT store per thread (b128 stores,
// write-once data kept out of L2).
// ---------------------------------------------------------------------------
__global__ __launch_bounds__(256)
void clim_splat_kernel(const float* __restrict__ clim_vec,  // [B*T*Cc]
                       float* __restrict__ out)             // [B*T*Cout*HW]
{
    const int bt = blockIdx.z;     // 0..191
    const int cc = blockIdx.y;     // 0..10
    const int e  = blockIdx.x * blockDim.x + threadIdx.x;  // v4f index in plane

    const float v = clim_vec[bt * kCc + cc];
    v4f v4 = {v, v, v, v};

    size_t plane = (size_t)bt * kCout + (kC + cc);
    v4f* dst = reinterpret_cast<v4f*>(out) + plane * (kHW / 4) + e;
    __builtin_nontemporal_store(v4, dst);
}

// ---------------------------------------------------------------------------
extern "C" void kernel_launch(void* const* d_in, const int* in_sizes, int n_in,
                              void* d_out, int out_size, void* d_ws, size_t ws_size,
                              hipStream_t stream)
{
    // setup_inputs() order: input_sat, dates_sat, input_clim, dates_clim
    const float* input_sat  = (const float*)d_in[0];
    const int*   dates_sat  = (const int*)  d_in[1];
    const float* input_clim = (const float*)d_in[2];
    const int*   dates_clim = (const int*)  d_in[3];
    float* out      = (float*)d_out;
    float* clim_vec = (float*)d_ws;   // needs B*T*Cc*4 = 8448 bytes

    (void)in_sizes; (void)n_in; (void)out_size; (void)ws_size;

    // 1) date match + gather (tiny)
    clim_match_kernel<<<1, 256, 0, stream>>>(dates_sat, input_clim, dates_clim, clim_vec);

    // 2) sat plane copies via async global<->LDS DMA path (no dep on kernel 1)
    sat_copy_async_kernel<<<kB * kT * kC, 256, 0, stream>>>(input_sat, out);

    // 3) climate broadcast planes (depends on kernel 1; stream order handles it)
    dim3 grid_splat(kHW / 4 / 256, kCc, kB * kT);
    clim_splat_kernel<<<grid_splat, 256, 0, stream>>>(clim_vec, out);
}